// StandardGNN_52063593562919
// MI455X (gfx1250) — compile-verified
//
#include <hip/hip_runtime.h>
#include <math.h>

#define HEADS 4
#define CH 32
#define HID 128
#define NEG_SLOPE 0.2f

typedef __attribute__((ext_vector_type(2))) float v2f;
typedef __attribute__((ext_vector_type(8))) float v8f;

// ---------------------------------------------------------------------------
// f32 WMMA GEMM: C[M,Nn] = A[M,K] @ B[K,Nn], all row-major, M%16==Nn%16==K%4==0.
// One wave32 per 16x16 output tile, K stepped by 4 via v_wmma_f32_16x16x4_f32.
// Fragment layouts per CDNA5 ISA 7.12.2:
//   A 16x4 : lane l holds A[l%16][2*(l/16)], A[l%16][2*(l/16)+1]
//   B 4x16 : lane l holds B[2*(l/16)][l%16], B[2*(l/16)+1][l%16]
//   C 16x16: vgpr r, lane l -> C[r + 8*(l/16)][l%16]
// ---------------------------------------------------------------------------
__global__ void k_gemm_f32_wmma(const float* __restrict__ A,
                                const float* __restrict__ B,
                                float* __restrict__ C,
                                int M, int Nn, int K) {
    const int wave = blockIdx.x * (blockDim.x >> 5) + (threadIdx.x >> 5);
    const int lane = threadIdx.x & 31;
    const int ntiles = Nn >> 4;
    const int mtiles = M >> 4;
    if (wave >= mtiles * ntiles) return;   // wave-uniform: EXEC stays all-1s
    const int tm = wave / ntiles;
    const int tn = wave - tm * ntiles;

    const int l16   = lane & 15;
    const int khalf = lane >> 4;           // 0 or 1
    const int arow  = tm * 16 + l16;       // A row for this lane
    const int bcol  = tn * 16 + l16;       // B col for this lane

    v8f acc = {};
    for (int k = 0; k < K; k += 4) {
        const int ka = k + 2 * khalf;
        v2f a, b;
        a.x = A[(size_t)arow * K + ka];
        a.y = A[(size_t)arow * K + ka + 1];
        b.x = B[(size_t)ka * Nn + bcol];
        b.y = B[(size_t)(ka + 1) * Nn + bcol];
        acc = __builtin_amdgcn_wmma_f32_16x16x4_f32(
            false, a, false, b, (short)0, acc, false, false);
    }
    const int crow = tm * 16 + 8 * khalf;
#pragma unroll
    for (int r = 0; r < 8; ++r)
        C[(size_t)(crow + r) * Nn + (tn * 16 + l16)] = acc[r];
}

// Layer 1: F_in == 1, so H[n,c] = x[n] * W1[c]
__global__ void k_layer1(const float* __restrict__ x, const float* __restrict__ W1,
                         float* __restrict__ H, int N) {
    const size_t i = (size_t)blockIdx.x * blockDim.x + threadIdx.x;
    if (i >= (size_t)N * HID) return;
    const int n = (int)(i >> 7);
    const int c = (int)(i & 127);
    H[i] = x[n] * W1[c];
}

// Per (node, head) attention coefficients: al_s/al_d [N,4]
__global__ void k_attn_coef(const float* __restrict__ H,
                            const float* __restrict__ a_src,
                            const float* __restrict__ a_dst,
                            float* __restrict__ al_s, float* __restrict__ al_d,
                            int N) {
    const int t = blockIdx.x * blockDim.x + threadIdx.x;  // t = n*4 + h
    if (t >= N * HEADS) return;
    const int n = t >> 2, h = t & 3;
    const float* hp = H + (size_t)n * HID + h * CH;
    const float* as = a_src + h * CH;
    const float* ad = a_dst + h * CH;
    float ss = 0.f, sd = 0.f;
#pragma unroll
    for (int c = 0; c < CH; ++c) { const float v = hp[c]; ss += v * as[c]; sd += v * ad[c]; }
    al_s[t] = ss;
    al_d[t] = sd;
}

__global__ void k_fill(float* __restrict__ p, float v, size_t n) {
    const size_t i = (size_t)blockIdx.x * blockDim.x + threadIdx.x;
    if (i < n) p[i] = v;
}

__device__ __forceinline__ float lrelu(float x) { return x > 0.f ? x : NEG_SLOPE * x; }

// Order-independent float atomic max via sign-split integer atomics.
// Requires initialization to -inf (0xFF800000).
__device__ __forceinline__ void atomicMaxFloat(float* addr, float val) {
    if (val >= 0.f) atomicMax((int*)addr, __float_as_int(val));
    else            atomicMin((unsigned int*)addr, (unsigned int)__float_as_int(val));
}

__device__ __forceinline__ void edge_sd(int e, int E, const int* __restrict__ srcE,
                                        const int* __restrict__ dstE, int& s, int& d) {
    if (e < E) { s = srcE[e]; d = dstE[e]; }
    else       { s = e - E;   d = s;       }   // appended self-loops
}

// Pass A: segment max of leaky_relu(al_s[src]+al_d[dst]) over dst
__global__ void k_edge_max(const float* __restrict__ al_s, const float* __restrict__ al_d,
                           const int* __restrict__ srcE, const int* __restrict__ dstE,
                           float* __restrict__ emax, int E, int Etot) {
    const int e = blockIdx.x * blockDim.x + threadIdx.x;
    if (e >= Etot) return;
    int s, d; edge_sd(e, E, srcE, dstE, s, d);
    const float4 as = *(const float4*)(al_s + (size_t)s * 4);
    const float4 ad = *(const float4*)(al_d + (size_t)d * 4);
    float* em = emax + (size_t)d * 4;
    atomicMaxFloat(em + 0, lrelu(as.x + ad.x));
    atomicMaxFloat(em + 1, lrelu(as.y + ad.y));
    atomicMaxFloat(em + 2, lrelu(as.z + ad.z));
    atomicMaxFloat(em + 3, lrelu(as.w + ad.w));
}

// Pass B: e_exp = exp(e - emax[dst]); denom = segment_sum(e_exp)
__global__ void k_edge_exp(const float* __restrict__ al_s, const float* __restrict__ al_d,
                           const float* __restrict__ emax, float* __restrict__ denom,
                           float* __restrict__ eexp,
                           const int* __restrict__ srcE, const int* __restrict__ dstE,
                           int E, int Etot) {
    const int e = blockIdx.x * blockDim.x + threadIdx.x;
    if (e >= Etot) return;
    int s, d; edge_sd(e, E, srcE, dstE, s, d);
    const float4 as = *(const float4*)(al_s + (size_t)s * 4);
    const float4 ad = *(const float4*)(al_d + (size_t)d * 4);
    const float4 em = *(const float4*)(emax + (size_t)d * 4);
    float4 ex;
    ex.x = __expf(lrelu(as.x + ad.x) - em.x);
    ex.y = __expf(lrelu(as.y + ad.y) - em.y);
    ex.z = __expf(lrelu(as.z + ad.z) - em.z);
    ex.w = __expf(lrelu(as.w + ad.w) - em.w);
    *(float4*)(eexp + (size_t)e * 4) = ex;
    float* dn = denom + (size_t)d * 4;
    atomicAdd(dn + 0, ex.x);
    atomicAdd(dn + 1, ex.y);
    atomicAdd(dn + 2, ex.z);
    atomicAdd(dn + 3, ex.w);
}

// Pass C (dominant traffic): one wave32 per edge; lane l moves channels 4l..4l+3
// (head = l/8), scaled by alpha, scatter-added to out[dst].
__global__ void k_edge_aggr(const float* __restrict__ H,
                            const float* __restrict__ eexp,
                            const float* __restrict__ denom,
                            const int* __restrict__ srcE, const int* __restrict__ dstE,
                            float* __restrict__ out, int E, int Etot) {
    const int wave = blockIdx.x * (blockDim.x >> 5) + (threadIdx.x >> 5);
    const int lane = threadIdx.x & 31;
    if (wave >= Etot) return;            // wave-uniform exit
    int s, d; edge_sd(wave, E, srcE, dstE, s, d);
    const int head = lane >> 3;
    const float alpha = eexp[(size_t)wave * 4 + head] / denom[(size_t)d * 4 + head];
    const float4 v = ((const float4*)(H + (size_t)s * HID))[lane];
    float* o = out + (size_t)d * HID + lane * 4;
    atomicAdd(o + 0, v.x * alpha);
    atomicAdd(o + 1, v.y * alpha);
    atomicAdd(o + 2, v.z * alpha);
    atomicAdd(o + 3, v.w * alpha);
}

__global__ void k_bias_relu(float* __restrict__ p, const float* __restrict__ b,
                            int N, int C) {
    const size_t i = (size_t)blockIdx.x * blockDim.x + threadIdx.x;
    if (i >= (size_t)N * C) return;
    const int c = (int)(i % C);
    const float v = p[i] + b[c];
    p[i] = v > 0.f ? v : 0.f;
}

// Final head: logits = mid[N,64] @ cw2[64,5] + cb2, then log_softmax
__global__ void k_head(const float* __restrict__ mid, const float* __restrict__ cw2,
                       const float* __restrict__ cb2, float* __restrict__ out, int N) {
    const int n = blockIdx.x * blockDim.x + threadIdx.x;
    if (n >= N) return;
    const float* m = mid + (size_t)n * 64;
    float lg[5];
#pragma unroll
    for (int j = 0; j < 5; ++j) {
        float sum = cb2[j];
        for (int c = 0; c < 64; ++c) sum += m[c] * cw2[c * 5 + j];
        lg[j] = sum;
    }
    float mx = lg[0];
#pragma unroll
    for (int j = 1; j < 5; ++j) mx = fmaxf(mx, lg[j]);
    float se = 0.f;
#pragma unroll
    for (int j = 0; j < 5; ++j) se += __expf(lg[j] - mx);
    const float lse = mx + __logf(se);
#pragma unroll
    for (int j = 0; j < 5; ++j) out[(size_t)n * 5 + j] = lg[j] - lse;
}

// ---------------------------------------------------------------------------

static inline int cdiv(long a, long b) { return (int)((a + b - 1) / b); }

extern "C" void kernel_launch(void* const* d_in, const int* in_sizes, int n_in,
                              void* d_out, int out_size, void* d_ws, size_t ws_size,
                              hipStream_t stream) {
    const float* x     = (const float*)d_in[0];
    const int*   ei    = (const int*)  d_in[1];
    const float* W1    = (const float*)d_in[2];
    const float* b1    = (const float*)d_in[3];
    const float* asrc1 = (const float*)d_in[4];
    const float* adst1 = (const float*)d_in[5];
    const float* W2    = (const float*)d_in[6];
    const float* b2    = (const float*)d_in[7];
    const float* asrc2 = (const float*)d_in[8];
    const float* adst2 = (const float*)d_in[9];
    const float* W3    = (const float*)d_in[10];
    const float* b3    = (const float*)d_in[11];
    const float* asrc3 = (const float*)d_in[12];
    const float* adst3 = (const float*)d_in[13];
    const float* cw1   = (const float*)d_in[14];
    const float* cb1   = (const float*)d_in[15];
    const float* cw2   = (const float*)d_in[16];
    const float* cb2   = (const float*)d_in[17];

    const int N    = in_sizes[0];        // x is [N,1]
    const int E    = in_sizes[1] / 2;    // edge_index [2,E]
    const int Etot = E + N;              // + self loops
    const int* srcE = ei;
    const int* dstE = ei + E;

    // Workspace carve-up (floats)
    float* ws = (float*)d_ws;
    size_t o = 0;
    float* bufA  = ws + o; o += (size_t)N * HID;
    float* bufB  = ws + o; o += (size_t)N * HID;
    float* bufC  = ws + o; o += (size_t)N * HID;
    float* al_s  = ws + o; o += (size_t)N * HEADS;
    float* al_d  = ws + o; o += (size_t)N * HEADS;
    float* emax  = ws + o; o += (size_t)N * HEADS;
    float* denom = ws + o; o += (size_t)N * HEADS;
    float* eexp  = ws + o; o += (size_t)Etot * HEADS;
    float* mid   = ws + o; o += (size_t)N * 64;
    (void)ws_size; (void)n_in; (void)out_size;

    const int T = 256;
    const float NEG_INF = -__builtin_huge_valf();

    // One GAT layer given H (=X@W already computed) accumulating into agg
    auto gat_attention = [&](const float* H, float* agg,
                             const float* asrc, const float* adst, const float* bias) {
        k_attn_coef<<<cdiv((long)N * HEADS, T), T, 0, stream>>>(H, asrc, adst, al_s, al_d, N);
        k_fill<<<cdiv((long)N * HEADS, T), T, 0, stream>>>(emax, NEG_INF, (size_t)N * HEADS);
        k_fill<<<cdiv((long)N * HEADS, T), T, 0, stream>>>(denom, 0.f, (size_t)N * HEADS);
        k_fill<<<cdiv((long)N * HID, T), T, 0, stream>>>(agg, 0.f, (size_t)N * HID);
        k_edge_max<<<cdiv(Etot, T), T, 0, stream>>>(al_s, al_d, srcE, dstE, emax, E, Etot);
        k_edge_exp<<<cdiv(Etot, T), T, 0, stream>>>(al_s, al_d, emax, denom, eexp,
                                                    srcE, dstE, E, Etot);
        k_edge_aggr<<<cdiv(Etot, 8), T, 0, stream>>>(H, eexp, denom, srcE, dstE,
                                                     agg, E, Etot);  // 8 waves/block
        k_bias_relu<<<cdiv((long)N * HID, T), T, 0, stream>>>(agg, bias, N, HID);
    };

    auto gemm = [&](const float* A, const float* B, float* C, int M, int Nn, int K) {
        const int waves = (M >> 4) * (Nn >> 4);
        k_gemm_f32_wmma<<<cdiv(waves, 4), 128, 0, stream>>>(A, B, C, M, Nn, K);
    };

    // Layer 1: F_in = 1, H in bufB, aggregate into bufC
    k_layer1<<<cdiv((long)N * HID, T), T, 0, stream>>>(x, W1, bufB, N);
    gat_attention(bufB, bufC, asrc1, adst1, b1);        // X2 = bufC

    // Layer 2: H = X2 @ W2 -> bufA, aggregate into bufB
    gemm(bufC, W2, bufA, N, HID, HID);
    gat_attention(bufA, bufB, asrc2, adst2, b2);        // X3 = bufB

    // Layer 3: H = X3 @ W3 -> bufC, aggregate into bufA
    gemm(bufB, W3, bufC, N, HID, HID);
    gat_attention(bufC, bufA, asrc3, adst3, b3);        // H3 = bufA

    // Classifier: mid = relu(H3 @ cw1 + cb1); logits -> log_softmax
    gemm(bufA, cw1, mid, N, 64, HID);
    k_bias_relu<<<cdiv((long)N * 64, T), T, 0, stream>>>(mid, cb1, N, 64);
    k_head<<<cdiv(N, T), T, 0, stream>>>(mid, cw2, cb2, (float*)d_out, N);
}